// TwoLayerModel_83167746719925
// MI455X (gfx1250) — compile-verified
//
#include <hip/hip_runtime.h>
#include <math.h>

// Parabolic erosion (min-plus) followed by parabolic dilation (max-plus),
// 7x7 'same' windows, fused in one kernel pass using separability of the
// parabolic structuring element (i^2 + j^2).
//
// x: [B, 64, 256, 256] f32, scale_min/scale_max: [64] f32, out: same as x.

#define HH   256
#define WW   256
#define CC   64
#define TILE 32
#define R1   3              // erosion/dilation window radius
#define R2   6              // fused halo radius
#define XW   (TILE + 2*R2)  // 44 : input halo tile width
#define EW   (TILE + 2*R1)  // 38 : erosion tile width

__global__ __launch_bounds__(256)
void parapool_fused(const float* __restrict__ x,
                    const float* __restrict__ smin,
                    const float* __restrict__ smax,
                    float* __restrict__ out)
{
    __shared__ float sx[XW * XW];    // 44x44 input halo tile      (7744 B)
    __shared__ float sv[EW * XW];    // 38x44 vertically eroded    (6688 B)
    __shared__ float se[EW * EW];    // 38x38 fully eroded         (5776 B)
    __shared__ float sd[TILE * EW];  // 32x38 vertically dilated   (4864 B)
    __shared__ float sdummy[1];      // sink for OOB async loads

    const int tid = threadIdx.x;                  // 0..255 (8 wave32)
    const int tx  = (blockIdx.x & 7) * TILE;      // tile origin x
    const int ty  = (blockIdx.x >> 3) * TILE;     // tile origin y
    const int c   = blockIdx.y;
    const int b   = blockIdx.z;

    const float sm = smin[c];
    const float sM = smax[c];
    float wm[7], wM[7];
#pragma unroll
    for (int i = 0; i < 7; ++i) {
        const float p = (float)((i - 3) * (i - 3));  // 9,4,1,0,1,4,9
        wm[i] = p * sm;
        wM[i] = p * sM;
    }

    const float* base = x + ((size_t)(b * CC + c)) * (HH * WW);

    // Border tiles: pre-fill the halo tile with +inf (semifield identity for
    // min-plus); async loads below overwrite only the in-image elements.
    const bool border = (ty < R2) || (tx < R2) ||
                        (ty + TILE + R2 > HH) || (tx + TILE + R2 > WW);
    if (border) {
        for (int it = tid; it < XW * XW; it += 256) sx[it] = INFINITY;
        __syncthreads();
    }

    // ---- Async global -> LDS staging of the 44x44 halo tile (ASYNCcnt) ----
    for (int it = tid; it < XW * XW; it += 256) {
        const int r   = it / XW;
        const int cc2 = it % XW;
        const int gy  = ty - R2 + r;
        const int gx  = tx - R2 + cc2;
        const bool ok = (gy >= 0) && (gy < HH) && (gx >= 0) && (gx < WW);
        // OOB lanes: keep EXEC uniform-safe by redirecting the LDS target to a
        // dummy word and clamping the global address into the image.
        const unsigned ldsoff = ok ? (unsigned)(size_t)&sx[it]
                                   : (unsigned)(size_t)&sdummy[0];
        const int cgy = gy < 0 ? 0 : (gy > HH - 1 ? HH - 1 : gy);
        const int cgx = gx < 0 ? 0 : (gx > WW - 1 ? WW - 1 : gx);
        const unsigned goff = (unsigned)((cgy << 8) + cgx) * 4u;
        asm volatile("global_load_async_to_lds_b32 %0, %1, %2"
                     :
                     : "v"(ldsoff), "v"(goff), "s"(base)
                     : "memory");
    }
    asm volatile("s_wait_asynccnt 0" ::: "memory");
    __syncthreads();

    // ---- Pass 1: vertical erosion  v[r][c] = min_i sx[r+i][c] + i^2*sm ----
    for (int it = tid; it < EW * XW; it += 256) {
        const int r   = it / XW;
        const int cc2 = it - r * XW;
        const float* col = &sx[r * XW + cc2];
        float m = col[0] + wm[0];
#pragma unroll
        for (int i = 1; i < 7; ++i) m = fminf(m, col[i * XW] + wm[i]);
        sv[it] = m;
    }
    __syncthreads();

    // ---- Pass 2: horizontal erosion + out-of-image mask (-inf) ----
    for (int it = tid; it < EW * EW; it += 256) {
        const int r  = it / EW;
        const int c2 = it - r * EW;
        const float* row = &sv[r * XW + c2];
        float m = row[0] + wm[0];
#pragma unroll
        for (int j = 1; j < 7; ++j) m = fminf(m, row[j] + wm[j]);
        const int gy = ty - R1 + r;
        const int gx = tx - R1 + c2;
        const bool valid = (gy >= 0) && (gy < HH) && (gx >= 0) && (gx < WW);
        se[it] = valid ? m : -INFINITY;   // dilation pads with -inf
    }
    __syncthreads();

    // ---- Pass 3: vertical dilation  d[r][c] = max_i e[r+i][c] - i^2*sM ----
    for (int it = tid; it < TILE * EW; it += 256) {
        const int r2 = it / EW;
        const int c2 = it - r2 * EW;
        const float* col = &se[r2 * EW + c2];
        float m = col[0] - wM[0];
#pragma unroll
        for (int i = 1; i < 7; ++i) m = fmaxf(m, col[i * EW] - wM[i]);
        sd[it] = m;
    }
    __syncthreads();

    // ---- Pass 4: horizontal dilation + streamed (NT) global store ----
    float* obase = out + ((size_t)(b * CC + c)) * (HH * WW);
    for (int it = tid; it < TILE * TILE; it += 256) {
        const int r2 = it >> 5;        // /32
        const int c3 = it & 31;        // %32
        const float* row = &sd[r2 * EW + c3];
        float m = row[0] - wM[0];
#pragma unroll
        for (int j = 1; j < 7; ++j) m = fmaxf(m, row[j] - wM[j]);
        __builtin_nontemporal_store(m, &obase[(size_t)(ty + r2) * WW + tx + c3]);
    }
}

extern "C" void kernel_launch(void* const* d_in, const int* in_sizes, int n_in,
                              void* d_out, int out_size, void* d_ws, size_t ws_size,
                              hipStream_t stream) {
    const float* x    = (const float*)d_in[0];
    const float* smin = (const float*)d_in[1];
    const float* smax = (const float*)d_in[2];
    float*       out  = (float*)d_out;

    const int B = in_sizes[0] / (CC * HH * WW);   // 8 for the reference shapes
    dim3 grid((HH / TILE) * (WW / TILE), CC, B);  // (64, 64, 8)
    parapool_fused<<<grid, 256, 0, stream>>>(x, smin, smax, out);
}